// BahdanauAttention_81449759802301
// MI455X (gfx1250) — compile-verified
//
#include <hip/hip_runtime.h>
#include <math.h>

typedef __attribute__((ext_vector_type(2))) float v2f;
typedef __attribute__((ext_vector_type(8))) float v8f;

static constexpr int kB = 4;
static constexpr int kS = 512;
static constexpr int kD = 256;
static constexpr int kU = 256;
static constexpr float kNEG = -1.0e9f;

// ---- hardware tanh (TRANS unit). Guaranteed v_tanh_f32 either way. --------
__device__ __forceinline__ float fast_tanhf(float x) {
#if __has_builtin(__builtin_amdgcn_tanhf)
    return __builtin_amdgcn_tanhf(x);
#elif __has_builtin(__builtin_amdgcn_tanh_f32)
    return __builtin_amdgcn_tanh_f32(x);
#else
    float y;
    // TRANS op: insert v_nop so the result is not consumed in the shadow
    // (ISA 7.3: 1 independent op or V_NOP after trans before output used).
    asm volatile("v_tanh_f32 %0, %1\n\tv_nop" : "=v"(y) : "v"(x));
    return y;
#endif
}

// ---- async global->LDS (ASYNCcnt path), guarded ---------------------------
#if __has_builtin(__builtin_amdgcn_global_load_async_to_lds_b64) && \
    __has_builtin(__builtin_amdgcn_global_load_async_to_lds_b128)
#define HAS_ASYNC_LDS 1
typedef int vi2 __attribute__((vector_size(8)));
typedef int vi4 __attribute__((vector_size(16)));
typedef __attribute__((address_space(1))) vi2* gp2_t;   // global v2i*
typedef __attribute__((address_space(3))) vi2* lp2_t;   // LDS v2i*
typedef __attribute__((address_space(1))) vi4* gp4_t;   // global v4i*
typedef __attribute__((address_space(3))) vi4* lp4_t;   // LDS v4i*
__device__ __forceinline__ void async_ld_b64(void* lds, const void* g) {
    __builtin_amdgcn_global_load_async_to_lds_b64((gp2_t)g, (lp2_t)lds, 0, 0);
}
__device__ __forceinline__ void async_ld_b128(void* lds, const void* g) {
    __builtin_amdgcn_global_load_async_to_lds_b128((gp4_t)g, (lp4_t)lds, 0, 0);
}
#else
#define HAS_ASYNC_LDS 0
#endif

__device__ __forceinline__ void wait_async0() {
#if __has_builtin(__builtin_amdgcn_s_wait_asynccnt)
    __builtin_amdgcn_s_wait_asynccnt(0);
#else
    asm volatile("s_wait_asynccnt 0" ::: "memory");
#endif
}

// f32 WMMA 16x16x4 lane layouts (CDNA5 ISA §7.12.2):
//   A (16x4):  lane l -> m = l%16 ; elem r -> k = (l/16)*2 + r
//   B (4x16):  lane l -> n = l%16 ; elem r -> k = (l/16)*2 + r
//   C/D(16x16):lane l -> n = l%16 ; elem r -> m = r + (l/16)*8

// ---------------------------------------------------------------------------
// K1: Q = X*Wq, V = X*Wv.  X:[2048,256] Wq/Wv:[256,256].
// One wave per 16x16 output tile; both Q and V tiles share the A tile.
// ---------------------------------------------------------------------------
__global__ __launch_bounds__(32)
void proj_qv_kernel(const float* __restrict__ X,
                    const float* __restrict__ Wq,
                    const float* __restrict__ Wv,
                    float* __restrict__ Q,
                    float* __restrict__ V) {
    const int nt = blockIdx.x & 15;       // 16 N tiles
    const int mt = blockIdx.x >> 4;       // 128 M tiles
    const int m0 = mt * 16, n0 = nt * 16;
    const int lane = threadIdx.x;

    __shared__ float Alds[16 * 260];      // 260%64==4 -> conflict-free
    __shared__ float Bq[256 * 16];        // [k][n]
    __shared__ float Bv[256 * 16];

    #pragma unroll
    for (int r = 0; r < 16; ++r) {
        const float4* src = (const float4*)(X + (size_t)(m0 + r) * kD);
        #pragma unroll
        for (int i = 0; i < 2; ++i)
            *(float4*)&Alds[r * 260 + (lane + i * 32) * 4] = src[lane + i * 32];
    }
    {
        const int r  = lane >> 2;
        const int c4 = (lane & 3) * 4;
        #pragma unroll 4
        for (int kb = 0; kb < 256; kb += 8) {
            float4 aq = *(const float4*)(Wq + (size_t)(kb + r) * kU + n0 + c4);
            float4 av = *(const float4*)(Wv + (size_t)(kb + r) * kU + n0 + c4);
            *(float4*)&Bq[(kb + r) * 16 + c4] = aq;
            *(float4*)&Bv[(kb + r) * 16 + c4] = av;
        }
    }
    __syncthreads();

    const int mn    = lane & 15;
    const int khalf = (lane >> 4) * 2;

    v8f cq = {};
    v8f cv = {};
    #pragma unroll 4
    for (int kb = 0; kb < 256; kb += 4) {
        const int k0 = kb + khalf;
        v2f a;  a.x  = Alds[mn * 260 + k0];  a.y  = Alds[mn * 260 + k0 + 1];
        v2f bq; bq.x = Bq[k0 * 16 + mn];     bq.y = Bq[(k0 + 1) * 16 + mn];
        v2f bv; bv.x = Bv[k0 * 16 + mn];     bv.y = Bv[(k0 + 1) * 16 + mn];
        cq = __builtin_amdgcn_wmma_f32_16x16x4_f32(false, a, false, bq,
                                                   (short)0, cq, false, false);
        cv = __builtin_amdgcn_wmma_f32_16x16x4_f32(false, a, false, bv,
                                                   (short)0, cv, false, false);
    }

    const int mrow0 = (lane >> 4) * 8;
    #pragma unroll
    for (int r = 0; r < 8; ++r) {
        const int row = m0 + mrow0 + r;
        Q[(size_t)row * kU + n0 + mn] = cq[r];
        V[(size_t)row * kU + n0 + mn] = cv[r];
    }
}

// ---------------------------------------------------------------------------
// K2: scores + causal/vmask + row softmax -> Attn [B,S,S].
// Block = 256 threads = 8 waves; block owns (b, 8 query rows i0..i0+7).
// v rows staged 32 at a time via async global->LDS (stride 258 -> 32 lanes
// hit 32 distinct banks when each lane walks its own row).
// ---------------------------------------------------------------------------
__global__ __launch_bounds__(256)
void score_softmax_kernel(const float* __restrict__ Q,
                          const float* __restrict__ Vp,
                          const float* __restrict__ Vw,
                          const unsigned char* __restrict__ mask,
                          float* __restrict__ Attn) {
    const int b    = blockIdx.x >> 6;         // 4 batches
    const int i0   = (blockIdx.x & 63) * 8;   // 64 row-groups
    const int tid  = threadIdx.x;
    const int w    = tid >> 5;
    const int lane = tid & 31;
    const int i    = i0 + w;

    __shared__ float qrow[8][256];
    __shared__ float vt[32 * 258];
    __shared__ float sc[8][512];
    __shared__ float vw[256];

    vw[tid] = Vw[tid];
    {   // 8 q rows = 2048 contiguous floats
        const float4* src = (const float4*)(Q + (size_t)(b * kS + i0) * kU);
        float4* dst = (float4*)&qrow[0][0];
        dst[tid]       = src[tid];
        dst[tid + 256] = src[tid + 256];
    }
    __syncthreads();

    const int njt = (i0 + 8 + 31) / 32;       // causal: tiles covering j<=i0+7
    const int row = tid >> 3, seg = tid & 7;  // staging role of this thread

    for (int jt = 0; jt < njt; ++jt) {
        const float* srcrow =
            Vp + (size_t)(b * kS + jt * 32 + row) * kU + seg * 32;
#if HAS_ASYNC_LDS
        // direct global->LDS, no VGPR bounce; 8B granule keeps stride-258
        #pragma unroll
        for (int q2 = 0; q2 < 16; ++q2)
            async_ld_b64(&vt[row * 258 + seg * 32 + q2 * 2], srcrow + q2 * 2);
        wait_async0();
#else
        #pragma unroll
        for (int q2 = 0; q2 < 16; ++q2)
            *(float2*)&vt[row * 258 + seg * 32 + q2 * 2] =
                *(const float2*)(srcrow + q2 * 2);
#endif
        __syncthreads();

        // overlap: prefetch next tile into L2/L0 while we crunch this one
        if (jt + 1 < njt)
            __builtin_prefetch(
                (const char*)(Vp + (size_t)(b * kS + (jt + 1) * 32) * kU) +
                    tid * 128, 0, 1);

        const int j = jt * 32 + lane;
        if (j < kS) {
            float s;
            if (j <= i) {
                float acc = 0.0f;
                const float* qr = &qrow[w][0];
                const float* vr = &vt[lane * 258];
                #pragma unroll 8
                for (int u = 0; u < 256; ++u)
                    acc += vw[u] * fast_tanhf(qr[u] + vr[u]);
                s = mask[b * kS + j] ? acc : acc + kNEG;
            } else {
                s = kNEG;                      // causal
            }
            sc[w][j] = s;
        }
        __syncthreads();                       // vt reused next tile
    }
    for (int j = njt * 32 + lane; j < kS; j += 32) sc[w][j] = kNEG;

    // per-wave softmax over row i (rows are wave-private)
    float mx = -3.0e38f;
    for (int j = lane; j < kS; j += 32) mx = fmaxf(mx, sc[w][j]);
    #pragma unroll
    for (int off = 16; off >= 1; off >>= 1) mx = fmaxf(mx, __shfl_xor(mx, off, 32));
    float sum = 0.0f;
    for (int j = lane; j < kS; j += 32) {
        float e = __expf(sc[w][j] - mx);       // exp(-1e9-mx) -> 0, as reference
        sc[w][j] = e;
        sum += e;
    }
    #pragma unroll
    for (int off = 16; off >= 1; off >>= 1) sum += __shfl_xor(sum, off, 32);
    const float inv = 1.0f / sum;

    float* dst = Attn + (size_t)(b * kS + i) * kS;
    for (int j = lane; j < kS; j += 32) dst[j] = sc[w][j] * inv;
}

// ---------------------------------------------------------------------------
// K3: context = Attn x values (per batch [512x512]*[512x256]), masked.
// One wave per 16x16 tile; K loop stops at m0+16 (attn is causal-zero beyond).
// ---------------------------------------------------------------------------
__global__ __launch_bounds__(32)
void context_kernel(const float* __restrict__ A,
                    const float* __restrict__ Vals,
                    const unsigned char* __restrict__ mask,
                    float* __restrict__ Out) {
    const int nt = blockIdx.x & 15;           // 16 N tiles
    const int mt = (blockIdx.x >> 4) & 31;    // 32 M tiles
    const int b  = blockIdx.x >> 9;           // 4 batches
    const int m0 = mt * 16, n0 = nt * 16;
    const int lane = threadIdx.x;
    const int kmax = m0 + 16;                 // causal cutoff

    __shared__ float Alds[16 * 516];          // 516%64==4, rows 16B-aligned

    for (int r = 0; r < 16; ++r) {
        const float* src = A + (size_t)(b * kS + m0 + r) * kS;
        for (int k = lane * 4; k < kmax; k += 128) {
#if HAS_ASYNC_LDS
            async_ld_b128(&Alds[r * 516 + k], src + k);
#else
            *(float4*)&Alds[r * 516 + k] = *(const float4*)&src[k];
#endif
        }
    }
#if HAS_ASYNC_LDS
    wait_async0();
#endif
    __syncthreads();

    const int mn    = lane & 15;
    const int khalf = (lane >> 4) * 2;

    v8f c = {};
    for (int kb = 0; kb < kmax; kb += 4) {
        const int k0 = kb + khalf;
        v2f a;  a.x = Alds[mn * 516 + k0];
                a.y = Alds[mn * 516 + k0 + 1];
        v2f bb; bb.x = Vals[(size_t)(b * kS + k0) * kD + n0 + mn];
                bb.y = Vals[(size_t)(b * kS + k0 + 1) * kD + n0 + mn];
        c = __builtin_amdgcn_wmma_f32_16x16x4_f32(false, a, false, bb,
                                                  (short)0, c, false, false);
    }

    const int mrow0 = (lane >> 4) * 8;
    #pragma unroll
    for (int r = 0; r < 8; ++r) {
        const int row  = m0 + mrow0 + r;
        const float mv = mask[b * kS + row] ? 1.0f : 0.0f;
        Out[(size_t)(b * kS + row) * kD + n0 + mn] = c[r] * mv;
    }
}

// ---------------------------------------------------------------------------
extern "C" void kernel_launch(void* const* d_in, const int* in_sizes, int n_in,
                              void* d_out, int out_size, void* d_ws, size_t ws_size,
                              hipStream_t stream) {
    (void)in_sizes; (void)n_in; (void)out_size; (void)ws_size;
    const float*         values = (const float*)d_in[0];        // [B,S,D]
    const unsigned char* mask   = (const unsigned char*)d_in[1];// [B,S] bool
    const float*         Wq     = (const float*)d_in[2];        // [D,U]
    const float*         Wv     = (const float*)d_in[3];        // [D,U]
    const float*         Vw     = (const float*)d_in[4];        // [U]
    float* out = (float*)d_out;

    float* q_ws    = (float*)d_ws;                       // [B*S,U]  2 MB
    float* v_ws    = q_ws + (size_t)kB * kS * kU;        // [B*S,U]  2 MB
    float* attn_ws = v_ws + (size_t)kB * kS * kU;        // [B,S,S]  4 MB

    proj_qv_kernel<<<(kB * kS / 16) * (kU / 16), 32, 0, stream>>>(
        values, Wq, Wv, q_ws, v_ws);
    score_softmax_kernel<<<kB * kS / 8, 256, 0, stream>>>(
        q_ws, v_ws, Vw, mask, attn_ws);
    context_kernel<<<kB * (kS / 16) * (kD / 16), 32, 0, stream>>>(
        attn_ws, values, mask, out);
}